// sideWinFiltering_20005957665145
// MI455X (gfx1250) — compile-verified
//
#include <hip/hip_runtime.h>

// Side-window filter, radius 3, 10 iterations, on [4,3,1024,1024] f32.
//
// MI455X reasoning:
//  * 48 MB tensor fits in 192 MB L2 -> the 10 iteration ping-pong chains
//    through L2; per-iter traffic ~96 MB is cheap even at HBM rate
//    (23.3 TB/s => ~41 us for all 10 iters). Compute is the co-dominant
//    cost, so use the separable decomposition (hl/hr 4-tap horizontal sums,
//    vertical prefix sums) = ~60 VALU ops/px instead of the naive
//    392-MAC/px conv. f32 WMMA (16x16x4 only) would be <6% utilized on the
//    band-matrix formulation and needs the undocumented B-operand layout,
//    so the matrix path is a pessimization for this stencil.
//  * gfx1250-specific path: halo tiles are pulled into LDS with
//    GLOBAL_LOAD_ASYNC_TO_LDS_B32 (ASYNCcnt) + s_wait_asynccnt — the CDNA5
//    async-copy engine, no VGPR round trip.
//  * wave32: 256-thread blocks = 8 waves; 32x32 tile; ~16 KB LDS/block so
//    many blocks co-resident per 320 KB WGP. hl/hr packed as float2 so
//    stage-2 uses ds_load_b64 and packed f32 ALU.

#define RADIUS   3
#define NITER    10
#define IMG_H    1024
#define IMG_W    1024
#define PLANES   12              // 4 batch * 3 channels, independent planes
#define TILE     32
#define INW      (TILE + 2*RADIUS)   // 38 rows/cols of tile+halo
#define TIN_STR  40                  // padded stride for input tile (floats)
#define HS_STR   33                  // stride for hl/hr array (float2 units)
#define NTHREADS 256

typedef __attribute__((ext_vector_type(2))) float f2;

__global__ __launch_bounds__(NTHREADS)
void swf_iter_kernel(const float* __restrict__ src, float* __restrict__ dst)
{
    __shared__ float s_tin[INW * TIN_STR];   // input tile + halo
    __shared__ f2    s_hlr[INW * HS_STR];    // {left,right} 4-tap horiz sums

    const int tid   = threadIdx.x;
    const int tx0   = blockIdx.x * TILE;
    const int ty0   = blockIdx.y * TILE;
    const int plane = blockIdx.z;
    const float* sp = src + (size_t)plane * (IMG_H * (size_t)IMG_W);

    // ---- Stage 0: async copy tile+halo (replicate-clamped) into LDS -------
    for (int idx = tid; idx < INW * INW; idx += NTHREADS) {
        int ly = idx / INW;
        int lx = idx - ly * INW;
        int gy = ty0 - RADIUS + ly; gy = gy < 0 ? 0 : (gy > IMG_H - 1 ? IMG_H - 1 : gy);
        int gx = tx0 - RADIUS + lx; gx = gx < 0 ? 0 : (gx > IMG_W - 1 ? IMG_W - 1 : gx);
        const float* gp = sp + gy * IMG_W + gx;
        // low 32 bits of the flat shared address == LDS byte offset
        unsigned lds_off = (unsigned)(size_t)(&s_tin[ly * TIN_STR + lx]);
        asm volatile("global_load_async_to_lds_b32 %0, %1, off"
                     :: "v"(lds_off), "v"(gp) : "memory");
    }
    asm volatile("s_wait_asynccnt 0x0" ::: "memory");
    __syncthreads();

    // ---- Stage 1: horizontal 4-tap sums for all 38 rows --------------------
    // hl[y][x] = sum in[y][x-3..x] ; hr[y][x] = sum in[y][x..x+3]
    for (int idx = tid; idx < INW * TILE; idx += NTHREADS) {
        int yi = idx >> 5;           // / 32
        int x  = idx & 31;
        const float* r = &s_tin[yi * TIN_STR + x];
        float t0 = r[0], t1 = r[1], t2 = r[2], t3 = r[3];
        float t4 = r[4], t5 = r[5], t6 = r[6];
        f2 v; v.x = t0 + t1 + t2 + t3; v.y = t3 + t4 + t5 + t6;
        s_hlr[yi * HS_STR + x] = v;
    }
    __syncthreads();

    // ---- Stage 2: vertical sums via prefix, 8 windows, argmin --------------
    const int x  = tid & 31;
    const int y0 = (tid >> 5) * 4;   // each thread: 4-pixel column, y0..y0+3

    f2    P[11];                      // prefix of {hl,hr} over 10 rows
    float Pc[11];                     // prefix of center column
    f2    c_h[10];
    float c_c[10];
    P[0] = (f2)(0.0f);
    Pc[0] = 0.0f;
#pragma unroll
    for (int i = 0; i < 10; ++i) {
        int yi = y0 + i;             // rows y0-3 .. y0+6 in image coords
        f2    h  = s_hlr[yi * HS_STR + x];      // ds_load_b64
        float cc = s_tin[yi * TIN_STR + x + RADIUS];
        c_h[i] = h; c_c[i] = cc;
        P [i + 1] = P [i] + h;
        Pc[i + 1] = Pc[i] + cc;
    }

    const float Qn = 1.0f / 16.0f;   // (r+1)^2  corner normalizer
    const float Hn = 1.0f / 28.0f;   // (r+1)*k  side   normalizer
    float* dp = dst + (size_t)plane * (IMG_H * (size_t)IMG_W);
    const unsigned obase = (unsigned)(ty0 + y0) * IMG_W + (unsigned)(tx0 + x);

#pragma unroll
    for (int j = 0; j < 4; ++j) {
        float xv   = c_c[j + 3];
        f2    vu   = P[j + 4] - P[j];           // rows y-3..y   {hl,hr}
        f2    vd   = P[j + 7] - P[j + 3];       // rows y..y+3   {hl,hr}
        float vu_c = Pc[j + 4] - Pc[j];
        float vd_c = Pc[j + 7] - Pc[j + 3];
        f2    ch   = c_h[j + 3];

        // directional mean - x, reference kernel order: L R U D NW NE SW SE
        float dL  = (vu.x + vd.x - ch.x) * Hn - xv;
        float dR  = (vu.y + vd.y - ch.y) * Hn - xv;
        float dU  = (vu.x + vu.y - vu_c) * Hn - xv;
        float dD  = (vd.x + vd.y - vd_c) * Hn - xv;
        float dNW = vu.x * Qn - xv;
        float dNE = vu.y * Qn - xv;
        float dSW = vd.x * Qn - xv;
        float dSE = vd.y * Qn - xv;

        float bd = dL, ba = fabsf(dL), a;       // first-min (argmin) semantics
        a = fabsf(dR);  if (a < ba) { ba = a; bd = dR;  }
        a = fabsf(dU);  if (a < ba) { ba = a; bd = dU;  }
        a = fabsf(dD);  if (a < ba) { ba = a; bd = dD;  }
        a = fabsf(dNW); if (a < ba) { ba = a; bd = dNW; }
        a = fabsf(dNE); if (a < ba) { ba = a; bd = dNE; }
        a = fabsf(dSW); if (a < ba) { ba = a; bd = dSW; }
        a = fabsf(dSE); if (a < ba) { ba = a; bd = dSE; }

        dp[obase + (unsigned)j * IMG_W] = xv + bd;
    }
}

extern "C" void kernel_launch(void* const* d_in, const int* in_sizes, int n_in,
                              void* d_out, int out_size, void* d_ws, size_t ws_size,
                              hipStream_t stream)
{
    (void)in_sizes; (void)n_in; (void)out_size; (void)ws_size;
    const float* img = (const float*)d_in[0];
    float* out = (float*)d_out;
    float* tmp = (float*)d_ws;           // needs PLANES*H*W*4 = 48 MB scratch

    dim3 grid(IMG_W / TILE, IMG_H / TILE, PLANES);
    dim3 block(NTHREADS);

    const float* src = img;
    for (int it = 0; it < NITER; ++it) {
        float* dst = (it & 1) ? out : tmp;   // it==9 (last) -> out
        swf_iter_kernel<<<grid, block, 0, stream>>>(src, dst);
        src = dst;
    }
}